// PolynomialLossNaive_83365315215382
// MI455X (gfx1250) — compile-verified
//
#include <hip/hip_runtime.h>
#include <hip/hip_bf16.h>

// loss = || F^T F - S^T S ||_F^2 / 2^38, with F^T = input viewed [64][8192] row-major.

typedef __attribute__((ext_vector_type(2))) float v2f;
typedef __attribute__((ext_vector_type(8))) float v8f;

#define CH 64
#define HW 8192
#define KCHUNK 128
#define NBLK (HW / KCHUNK)   // 64 blocks

// Pass 1: each wave computes one 16x16 tile of the partial (F^T F - S^T S)
// over this block's K-chunk using V_WMMA_F32_16X16X4_F32, then atomically
// accumulates into gdiff[64*64].
__global__ __launch_bounds__(512)
void gram_diff_kernel(const float* __restrict__ F,
                      const float* __restrict__ S,
                      float* __restrict__ gdiff) {
  const int wave = threadIdx.x >> 5;   // 0..15 -> one 16x16 output tile each
  const int lane = threadIdx.x & 31;
  const int m    = wave >> 2;          // row tile 0..3
  const int n    = wave & 3;           // col tile 0..3
  const int half = lane >> 4;          // 0: K=0,1  1: K=2,3 (A and B share striping)
  const int l15  = lane & 15;

  const int mrow = 16 * m + l15;       // A-fragment row (M index)
  const int nrow = 16 * n + l15;       // B-fragment column (N index)
  const int k0   = blockIdx.x * KCHUNK;

  const float* __restrict__ Fa = F + (size_t)mrow * HW;
  const float* __restrict__ Fb = F + (size_t)nrow * HW;
  const float* __restrict__ Sa = S + (size_t)mrow * HW;
  const float* __restrict__ Sb = S + (size_t)nrow * HW;

  v8f accF = {};
  v8f accS = {};

#pragma unroll 4
  for (int k = k0; k < k0 + KCHUNK; k += 4) {
    const int off = k + 2 * half;
    v2f fa = *(const v2f*)(Fa + off);
    v2f fb = *(const v2f*)(Fb + off);
    v2f sa = *(const v2f*)(Sa + off);
    v2f sb = *(const v2f*)(Sb + off);
    // 8 args: (neg_a, A, neg_b, B, c_mod, C, reuse_a, reuse_b)
    accF = __builtin_amdgcn_wmma_f32_16x16x4_f32(false, fa, false, fb,
                                                 (short)0, accF, false, false);
    accS = __builtin_amdgcn_wmma_f32_16x16x4_f32(false, sa, false, sb,
                                                 (short)0, accS, false, false);
  }

  // C/D layout: VGPR j holds (M = j + 8*half, N = l15) for this tile.
#pragma unroll
  for (int j = 0; j < 8; ++j) {
    const int row = 16 * m + j + 8 * half;
    const int col = 16 * n + l15;
    atomicAdd(&gdiff[row * CH + col], accF[j] - accS[j]);
  }
}

// Pass 2: sum of squares of the 64x64 difference Gram, scaled by 2^-38.
__global__ __launch_bounds__(256)
void gram_reduce_kernel(const float* __restrict__ gdiff, float* __restrict__ out) {
  __shared__ float red[256];
  float s = 0.0f;
  for (int i = threadIdx.x; i < CH * CH; i += 256) {
    const float v = gdiff[i];
    s += v * v;
  }
  red[threadIdx.x] = s;
  __syncthreads();
  for (int w = 128; w > 0; w >>= 1) {
    if (threadIdx.x < w) red[threadIdx.x] += red[threadIdx.x + w];
    __syncthreads();
  }
  if (threadIdx.x == 0) {
    // denom = hw_in * hw_tgt * ch^2 = 8192*8192*4096 = 2^38 (exact power of two)
    out[0] = red[0] * (1.0f / 274877906944.0f);
  }
}

extern "C" void kernel_launch(void* const* d_in, const int* in_sizes, int n_in,
                              void* d_out, int out_size, void* d_ws, size_t ws_size,
                              hipStream_t stream) {
  (void)in_sizes; (void)n_in; (void)out_size; (void)ws_size;
  const float* F = (const float*)d_in[0];   // input  -> F^T [64][8192]
  const float* S = (const float*)d_in[1];   // target -> S^T [64][8192]
  float* gdiff   = (float*)d_ws;            // 64*64 floats of scratch
  float* out     = (float*)d_out;

  hipMemsetAsync(gdiff, 0, CH * CH * sizeof(float), stream);
  gram_diff_kernel<<<NBLK, 512, 0, stream>>>(F, S, gdiff);
  gram_reduce_kernel<<<1, 256, 0, stream>>>(gdiff, out);
}